// CRFDependency_4733053960797
// MI455X (gfx1250) — compile-verified
//
#include <hip/hip_runtime.h>

// CRF dependency (Eisner inside) loss for B=256, N=160 on gfx1250 / MI455X.
// - DP tables laid out [h][d][b] (batch fastest): every wave32 access coalesced.
// - Working set (arc + s_i + s_c = ~79MB) is L2-resident (192MB L2); HBM is
//   touched only for the initial 26MB scores read -> the DP is L2-latency bound.
// - No matmul content: the (logsumexp,+) semiring over diagonal stripes has no
//   shared contraction dim, so WMMA can't help; the win is wave32 VALU with
//   native v_exp_f32/v_log_f32 plus CDNA5 async-to-LDS staging for latency
//   hiding (global_load_async_to_lds_b32 + s_wait_asynccnt).
// - One kernel launch per span width: the only same-step s_i dependencies are
//   produced by the same block and kept in registers; stale reads of those two
//   cells are -inf (init value) and are skipped by the online LSE, so all three
//   reductions fuse into ONE loop -> 3x memory-level parallelism.

#define Bsz   256
#define Ns    160
#define NNB   ((size_t)Ns * (size_t)Ns * (size_t)Bsz)
#define NB    ((size_t)Ns * (size_t)Bsz)
#define NEG_INF (-__builtin_inff())
#define CHUNK 4

#if __has_builtin(__builtin_amdgcn_global_load_async_to_lds_b32) && \
    __has_builtin(__builtin_amdgcn_s_wait_asynccnt)
#define ASYNC_LDS 1
// Builtin wants AS1 (global) int* / AS3 (LDS) int*. Build them via integer
// casts (generic LDS ptr low 32 bits == LDS offset; generic global == AS1).
typedef __attribute__((address_space(1))) int g_as1_i32;
typedef __attribute__((address_space(3))) int l_as3_i32;
#define TO_GLB(p) ((g_as1_i32*)(unsigned long long)(p))
#define TO_LDS(p) ((l_as3_i32*)(unsigned)(unsigned long long)(p))
#else
#define ASYNC_LDS 0
#endif

__device__ __forceinline__ void lse_acc(float x, float& m, float& s) {
    // Online logsumexp accumulate; skips -inf terms (matches the safe _lse).
    if (x > m) {
        s = s * __expf(m - x) + 1.0f;
        m = x;
    } else if (x > NEG_INF) {
        s += __expf(x - m);
    }
}

__device__ __forceinline__ float lse_fin(float m, float s) {
    return (s > 0.0f) ? (m + __logf(s)) : NEG_INF;
}

// ---------------------------------------------------------------------------
// Init: transpose scores [B,N,N] -> arc [N,N,B]; s_i = -inf; s_c = -inf except
// diagonal = 0; lens[b] = sum(mask[b,:]). All 32-bit index math.
// ---------------------------------------------------------------------------
__global__ void eisner_init(const float* __restrict__ scores,
                            const unsigned char* __restrict__ mask,
                            float* __restrict__ arc,
                            float* __restrict__ s_i,
                            float* __restrict__ s_c,
                            int* __restrict__ lens) {
    unsigned t = blockIdx.x * 256u + threadIdx.x;  // t in [0, NNB), fits u32
    unsigned b = t & 255u;
    unsigned r = t >> 8;          // r < N*N = 25600
    unsigned d = r % (unsigned)Ns;
    unsigned h = r / (unsigned)Ns;
    arc[t] = scores[(b * (unsigned)Ns + h) * (unsigned)Ns + d];
    s_i[t] = NEG_INF;
    s_c[t] = (h == d) ? 0.0f : NEG_INF;
    if (t < (unsigned)Bsz) {
        int L = 0;
        for (int i = 0; i < Ns; ++i) L += mask[t * (unsigned)Ns + i] ? 1 : 0;
        lens[t] = L;
    }
}

// ---------------------------------------------------------------------------
// One DP step at width w. Block k = span (k, k+w), thread = batch b.
// Fused over j = 0..w-1:
//   il: s_c[k,k+j]     + s_c[k+w,k+j+1]   (both contiguous, staged via async)
//   cl: s_i[k+w,k+j]   + s_c[k+j,k]       (row staged; column strided)
//   cr: s_i[k,k+j+1]   + s_c[k+j+1,k+w]   (row staged; column strided)
// Boundary terms (j=0 of cl, j=w-1 of cr) read the not-yet-written s_i cells,
// which are -inf since init -> skipped; the true values (siL/siR) are merged
// from registers after the loop. No intra-step races anywhere.
// ---------------------------------------------------------------------------
__global__ void eisner_step(float* __restrict__ s_i,
                            float* __restrict__ s_c,
                            const float* __restrict__ arc,
                            const int* __restrict__ lens,
                            int w) {
    const int k = blockIdx.x;
    const int b = threadIdx.x;

    const float* pIL  = s_c + ((size_t)k * Ns + k) * Bsz + b;             // stride Bsz
    const float* pIR  = s_c + ((size_t)(k + w) * Ns + (k + 1)) * Bsz + b; // stride Bsz
    const float* pCLi = s_i + ((size_t)(k + w) * Ns + k) * Bsz + b;       // stride Bsz
    const float* pCRi = s_i + ((size_t)k * Ns + (k + 1)) * Bsz + b;       // stride Bsz
    const float* pCLc = s_c + ((size_t)k * Ns + k) * Bsz + b;             // stride NB
    const float* pCRc = s_c + ((size_t)(k + 1) * Ns + (k + w)) * Bsz + b; // stride NB

    float mI = NEG_INF, sI = 0.0f;
    float mL = NEG_INF, sL = 0.0f;
    float mR = NEG_INF, sR = 0.0f;

#if ASYNC_LDS
    __shared__ float bIL[2][CHUNK][Bsz];
    __shared__ float bIR[2][CHUNK][Bsz];
    __shared__ float bCL[2][CHUNK][Bsz];
    __shared__ float bCR[2][CHUNK][Bsz];

    // Prime chunk 0.
    {
        int cn = (w < CHUNK) ? w : CHUNK;
        for (int j = 0; j < cn; ++j) {
            size_t o = (size_t)j * Bsz;
            __builtin_amdgcn_global_load_async_to_lds_b32(TO_GLB(pIL  + o), TO_LDS(&bIL[0][j][b]), 0, 0);
            __builtin_amdgcn_global_load_async_to_lds_b32(TO_GLB(pIR  + o), TO_LDS(&bIR[0][j][b]), 0, 0);
            __builtin_amdgcn_global_load_async_to_lds_b32(TO_GLB(pCLi + o), TO_LDS(&bCL[0][j][b]), 0, 0);
            __builtin_amdgcn_global_load_async_to_lds_b32(TO_GLB(pCRi + o), TO_LDS(&bCR[0][j][b]), 0, 0);
        }
    }
    int cur = 0;
    for (int j0 = 0; j0 < w; j0 += CHUNK, cur ^= 1) {
        __builtin_amdgcn_s_wait_asynccnt(0);  // current chunk resident in LDS
        int nj0 = j0 + CHUNK;
        if (nj0 < w) {  // prefetch next chunk, overlapped with compute below
            int cn = (w - nj0 < CHUNK) ? (w - nj0) : CHUNK;
            for (int j = 0; j < cn; ++j) {
                size_t o = (size_t)(nj0 + j) * Bsz;
                __builtin_amdgcn_global_load_async_to_lds_b32(TO_GLB(pIL  + o), TO_LDS(&bIL[cur ^ 1][j][b]), 0, 0);
                __builtin_amdgcn_global_load_async_to_lds_b32(TO_GLB(pIR  + o), TO_LDS(&bIR[cur ^ 1][j][b]), 0, 0);
                __builtin_amdgcn_global_load_async_to_lds_b32(TO_GLB(pCLi + o), TO_LDS(&bCL[cur ^ 1][j][b]), 0, 0);
                __builtin_amdgcn_global_load_async_to_lds_b32(TO_GLB(pCRi + o), TO_LDS(&bCR[cur ^ 1][j][b]), 0, 0);
            }
        }
        int cc = (w - j0 < CHUNK) ? (w - j0) : CHUNK;
        for (int j = 0; j < cc; ++j) {
            int jg = j0 + j;
            if (jg + CHUNK < w) {
                __builtin_prefetch((const void*)(pCLc + (size_t)(jg + CHUNK) * NB), 0, 1);
                __builtin_prefetch((const void*)(pCRc + (size_t)(jg + CHUNK) * NB), 0, 1);
            }
            float clc = pCLc[(size_t)jg * NB];  // strided column loads
            float crc = pCRc[(size_t)jg * NB];
            lse_acc(bIL[cur][j][b] + bIR[cur][j][b], mI, sI);
            lse_acc(bCL[cur][j][b] + clc,            mL, sL);
            lse_acc(bCR[cur][j][b] + crc,            mR, sR);
        }
    }
#else
#pragma unroll 2
    for (int j = 0; j < w; ++j) {
        size_t o = (size_t)j * Bsz;
        if (j + 4 < w) {
            __builtin_prefetch((const void*)(pCLc + (size_t)(j + 4) * NB), 0, 1);
            __builtin_prefetch((const void*)(pCRc + (size_t)(j + 4) * NB), 0, 1);
        }
        lse_acc(pIL[o]  + pIR[o],               mI, sI);
        lse_acc(pCLi[o] + pCLc[(size_t)j * NB], mL, sL);
        lse_acc(pCRi[o] + pCRc[(size_t)j * NB], mR, sR);
    }
#endif

    float il = lse_fin(mI, sI);

    const size_t idxL = ((size_t)(k + w) * Ns + k) * Bsz + b;  // [k+w, k]
    const size_t idxR = ((size_t)k * Ns + (k + w)) * Bsz + b;  // [k, k+w]
    float siL = il + arc[idxL];
    float siR = il + arc[idxR];
    s_i[idxL] = siL;
    s_i[idxR] = siR;

    lse_acc(siL, mL, sL);  // true j=0 term of cl (s_c[k,k] = 0)
    lse_acc(siR, mR, sR);  // true j=w-1 term of cr (s_c[k+w,k+w] = 0)
    float clv = lse_fin(mL, sL);
    float crv = lse_fin(mR, sR);

    s_c[idxL] = clv;
    if (k == 0) crv = (lens[b] == w) ? crv : NEG_INF;  // single-root constraint
    s_c[idxR] = crv;
}

// ---------------------------------------------------------------------------
// Per-batch gold score + root inside score (deterministic LDS tree reduce).
// ---------------------------------------------------------------------------
__global__ void eisner_gold(const float* __restrict__ scores,
                            const unsigned char* __restrict__ mask,
                            const int* __restrict__ target,
                            const float* __restrict__ s_c,
                            const int* __restrict__ lens,
                            float* __restrict__ partials) {
    int b = blockIdx.x;
    int t = threadIdx.x;
    __shared__ float red[256];
    float acc = 0.0f;
    for (int i = t; i < Ns; i += 256) {
        if (mask[(size_t)b * Ns + i])
            acc += scores[((size_t)b * Ns + i) * Ns + target[(size_t)b * Ns + i]];
    }
    red[t] = acc;
    __syncthreads();
    for (int off = 128; off > 0; off >>= 1) {
        if (t < off) red[t] += red[t + off];
        __syncthreads();
    }
    if (t == 0) {
        partials[b] = red[0];
        partials[Bsz + b] = s_c[(size_t)lens[b] * Bsz + b];  // s_c[0, lens[b], b]
    }
}

__global__ void eisner_finalize(const float* __restrict__ partials,
                                const int* __restrict__ lens,
                                float* __restrict__ out) {
    __shared__ float g[256];
    __shared__ float r[256];
    __shared__ int   L[256];
    int t = threadIdx.x;
    g[t] = partials[t];
    r[t] = partials[Bsz + t];
    L[t] = lens[t];
    __syncthreads();
    for (int off = 128; off > 0; off >>= 1) {
        if (t < off) { g[t] += g[t + off]; r[t] += r[t + off]; L[t] += L[t + off]; }
        __syncthreads();
    }
    if (t == 0) out[0] = (r[0] - g[0]) / (float)L[0];
}

extern "C" void kernel_launch(void* const* d_in, const int* in_sizes, int n_in,
                              void* d_out, int out_size, void* d_ws, size_t ws_size,
                              hipStream_t stream) {
    const float*         scores = (const float*)d_in[0];         // [B,N,N] f32
    const unsigned char* mask   = (const unsigned char*)d_in[1]; // [B,N] bool (1B)
    const int*           target = (const int*)d_in[2];           // [B,N] i32

    float* arc      = (float*)d_ws;          // [N,N,B]
    float* s_i      = arc + NNB;             // [N,N,B]
    float* s_c      = s_i + NNB;             // [N,N,B]
    float* partials = s_c + NNB;             // [2*B]
    int*   lens     = (int*)(partials + 2 * Bsz);  // [B]

    eisner_init<<<(unsigned)(NNB / 256), 256, 0, stream>>>(scores, mask, arc, s_i, s_c, lens);

    for (int w = 1; w < Ns; ++w)
        eisner_step<<<Ns - w, Bsz, 0, stream>>>(s_i, s_c, arc, lens, w);

    eisner_gold<<<Bsz, 256, 0, stream>>>(scores, mask, target, s_c, lens, partials);
    eisner_finalize<<<1, 256, 0, stream>>>(partials, lens, (float*)d_out);
}